// GNNAutoEncoder_42279658062121
// MI455X (gfx1250) — compile-verified
//
#include <hip/hip_runtime.h>
#include <hip/hip_bf16.h>

typedef __attribute__((ext_vector_type(2))) float v2f;
typedef __attribute__((ext_vector_type(8))) float v8f;

__global__ void zero_kernel(float* __restrict__ p, int n) {
  int i = blockIdx.x * blockDim.x + threadIdx.x;
  if (i < n) p[i] = 0.0f;
}

// One thread per edge: agg[dst] += feat[src]; optionally count degrees.
// Source row loads vectorized (rows of D=12/24 floats are 16B aligned,
// D=6 rows are 8B aligned). Atomic destinations are L2-resident (<=9.6MB).
template<int D, bool COUNT>
__global__ void scatter_mean_kernel(const int* __restrict__ ei, int E,
                                    const float* __restrict__ feat,
                                    float* __restrict__ agg,
                                    float* __restrict__ cnt) {
  int e = blockIdx.x * blockDim.x + threadIdx.x;
  if (e >= E) return;
  int s = ei[e];
  int d = ei[E + e];

  float fv[D];
  if constexpr (D % 4 == 0) {
    const float4* fr = (const float4*)(feat + (size_t)s * D);
#pragma unroll
    for (int q = 0; q < D / 4; ++q) {
      float4 t = fr[q];
      fv[4 * q + 0] = t.x; fv[4 * q + 1] = t.y;
      fv[4 * q + 2] = t.z; fv[4 * q + 3] = t.w;
    }
  } else {
    const float2* fr = (const float2*)(feat + (size_t)s * D);
#pragma unroll
    for (int q = 0; q < D / 2; ++q) {
      float2 t = fr[q];
      fv[2 * q + 0] = t.x; fv[2 * q + 1] = t.y;
    }
  }

  float* ar = agg + (size_t)d * D;
#pragma unroll
  for (int f = 0; f < D; ++f) atomicAdd(&ar[f], fv[f]);
  if (COUNT) atomicAdd(&cnt[d], 1.0f);
}

// out[n, D_OUT] = (agg[n]/max(cnt,1)) @ Wl^T + bias + xin[n] @ Wr^T  (+ReLU)
// One wave per 16-node tile; fp32 V_WMMA_F32_16X16X4_F32, K = 2*D_IN fused
// as A = [mean | x], B = [Wl^T ; Wr^T]. Bias folded into C init.
// All loads use clamped indices -> unconditionally safe -> the compiler emits
// v_cndmask selects instead of exec-mask branches between WMMAs.
template<int D_IN, int D_OUT, bool RELU>
__global__ __launch_bounds__(256) void sage_layer_kernel(
    const float* __restrict__ agg, const float* __restrict__ cnt,
    const float* __restrict__ xin,
    const float* __restrict__ Wl, const float* __restrict__ Wr,
    const float* __restrict__ bias,
    float* __restrict__ out, int nNodes) {
  constexpr int KTOT   = 2 * D_IN;     // 12 / 24 / 48 -> multiple of 4
  constexpr int KSTEPS = KTOT / 4;
  constexpr int NTILES = (D_OUT + 15) / 16;

  const int wave = (int)((blockIdx.x * blockDim.x + threadIdx.x) >> 5);
  if (wave * 16 >= nNodes) return;     // wave-uniform exit: EXEC all-1s below
  const int lane = (int)(threadIdx.x & 31);
  const int half = lane >> 4;          // 0: lanes 0-15, 1: lanes 16-31
  const int row  = lane & 15;

  const int anode = min(wave * 16 + row, nNodes - 1);
  const float invc = 1.0f / fmaxf(cnt[anode], 1.0f);

  // A fragment (ISA 32-bit 16x4 layout): lane holds A[row][k0], A[row][k0+1],
  // k0 = 4*step + 2*half. Element k<D_IN comes from mean, else from xin.
  float afrag[KSTEPS][2];
#pragma unroll
  for (int s = 0; s < KSTEPS; ++s) {
#pragma unroll
    for (int j = 0; j < 2; ++j) {
      const int  k    = 4 * s + 2 * half + j;
      const bool isM  = (k < D_IN);
      const int  kk   = isM ? k : k - D_IN;         // always in [0, D_IN)
      const float* Ap = isM ? agg : xin;            // pointer select, no branch
      float v = Ap[anode * D_IN + kk];
      afrag[s][j] = isM ? v * invc : v;
    }
  }

#pragma unroll
  for (int nt = 0; nt < NTILES; ++nt) {
    const int   col   = nt * 16 + row;   // B/C/D column owned by this lane
    const bool  cok   = (col < D_OUT);
    const int   colc  = min(col, D_OUT - 1);        // safe index
    const float cmask = cok ? 1.0f : 0.0f;
    const float bc    = bias[colc] * cmask;
    v8f acc;
#pragma unroll
    for (int v = 0; v < 8; ++v) acc[v] = bc;        // C[m][col] = bias[col]

#pragma unroll
    for (int s = 0; s < KSTEPS; ++s) {
      v2f a, b;
      a.x = afrag[s][0];
      a.y = afrag[s][1];
#pragma unroll
      for (int j = 0; j < 2; ++j) {
        const int  k    = 4 * s + 2 * half + j;
        const bool isL  = (k < D_IN);
        const int  kk   = isL ? k : k - D_IN;
        const float* Wp = isL ? Wl : Wr;            // pointer select, no branch
        const float w   = Wp[colc * D_IN + kk] * cmask;
        if (j == 0) b.x = w; else b.y = w;
      }
      acc = __builtin_amdgcn_wmma_f32_16x16x4_f32(false, a, false, b,
                                                  (short)0, acc, false, false);
    }

#pragma unroll
    for (int v = 0; v < 8; ++v) {
      const int m = wave * 16 + v + 8 * half;       // D row for this VGPR
      if (cok && m < nNodes) {
        float r = acc[v];
        if (RELU) r = fmaxf(r, 0.0f);
        out[m * D_OUT + col] = r;
      }
    }
  }
}

extern "C" void kernel_launch(void* const* d_in, const int* in_sizes, int n_in,
                              void* d_out, int out_size, void* d_ws, size_t ws_size,
                              hipStream_t stream) {
  const float* x   = (const float*)d_in[0];
  const int*   ei  = (const int*)d_in[1];
  const float* W1l = (const float*)d_in[2];
  const float* b1  = (const float*)d_in[3];
  const float* W1r = (const float*)d_in[4];
  const float* W2l = (const float*)d_in[5];
  const float* b2  = (const float*)d_in[6];
  const float* W2r = (const float*)d_in[7];
  const float* W3l = (const float*)d_in[8];
  const float* b3  = (const float*)d_in[9];
  const float* W3r = (const float*)d_in[10];
  float* z = (float*)d_out;

  const int N = in_sizes[0] / 6;
  const int E = in_sizes[1] / 2;

  // Workspace layout (floats): zero-needed buffers first (43N), then h1/h2.
  float* ws   = (float*)d_ws;
  float* cnt  = ws;               // N
  float* agg1 = cnt  + N;         // 6N
  float* agg2 = agg1 + 6 * N;     // 12N
  float* agg3 = agg2 + 12 * N;    // 24N
  float* h1   = agg3 + 24 * N;    // 12N
  float* h2   = h1   + 12 * N;    // 24N   (total 79N floats ~ 31.6 MB)

  const int nzero = 43 * N;
  zero_kernel<<<(nzero + 255) / 256, 256, 0, stream>>>(ws, nzero);

  const int eb    = (E + 255) / 256;
  const int tiles = (N + 15) / 16;
  const int lb    = (tiles + 7) / 8;   // 8 waves (16-node tiles) per block

  // Layer 1: d 6 -> 12 (degree count fused into first scatter)
  scatter_mean_kernel<6, true><<<eb, 256, 0, stream>>>(ei, E, x, agg1, cnt);
  sage_layer_kernel<6, 12, true><<<lb, 256, 0, stream>>>(agg1, cnt, x,
                                                         W1l, W1r, b1, h1, N);
  // Layer 2: d 12 -> 24
  scatter_mean_kernel<12, false><<<eb, 256, 0, stream>>>(ei, E, h1, agg2, nullptr);
  sage_layer_kernel<12, 24, true><<<lb, 256, 0, stream>>>(agg2, cnt, h1,
                                                          W2l, W2r, b2, h2, N);
  // Layer 3: d 24 -> 6 (no ReLU), straight to output
  scatter_mean_kernel<24, false><<<eb, 256, 0, stream>>>(ei, E, h2, agg3, nullptr);
  sage_layer_kernel<24, 6, false><<<lb, 256, 0, stream>>>(agg3, cnt, h2,
                                                          W3l, W3r, b3, z, N);
}